// _YOLOTarget_17643725652510
// MI455X (gfx1250) — compile-verified
//
#include <hip/hip_runtime.h>

// ---------------------------------------------------------------------------
// YOLO target loss: loss = sum_i [post[i,4] >= 0.25] * (post[i,5] + sum(boxes[i,:]))
// fallback: if loss == 0.0 -> post[:,4].sum() * 0.0
// Pure HBM-bandwidth-bound streaming reduction (~160MB @ 23.3TB/s ~= 7us).
// Wave-level reduction uses V_WMMA_F32_16X16X4_F32 (full f32 precision).
// ---------------------------------------------------------------------------

typedef __attribute__((ext_vector_type(2))) float v2f;
typedef __attribute__((ext_vector_type(8))) float v8f;

#define CONF_THRES 0.25f
#define NBLK 1024
#define TPB  256
#define NWAVES (TPB / 32)

// Wave32 sum of (x0 + x1) across all lanes, via V_WMMA_F32_16X16X4_F32.
// A matrix (16x4 f32): lane l supplies cells (M = l&15, K = 2*(l>=16)+{0,1})
// from {x0, x1}; B = ones(4x16); D[m][n] = rowsum_m(A).
// D layout: vgpr v, lanes 0-15 -> M=v; lanes 16-31 -> M=v+8. So
// sum(d[0..7]) on lane 0 = rows 0..7, on lane 16 = rows 8..15.
__device__ __forceinline__ float wave_sum_f32_wmma(float x0, float x1) {
    v2f a; a[0] = x0;  a[1] = x1;
    v2f b; b[0] = 1.0f; b[1] = 1.0f;
    v8f c = {};
    v8f d = __builtin_amdgcn_wmma_f32_16x16x4_f32(
        /*neg_a=*/false, a, /*neg_b=*/false, b,
        /*c_mod=*/(short)0, c, /*reuse_a=*/false, /*reuse_b=*/false);
    float s = d[0] + d[1] + d[2] + d[3] + d[4] + d[5] + d[6] + d[7];
    float lo = __int_as_float(__builtin_amdgcn_readlane(__float_as_int(s), 0));
    float hi = __int_as_float(__builtin_amdgcn_readlane(__float_as_int(s), 16));
    return lo + hi;
}

// Pass 1: grid-stride streaming accumulate -> one (loss, s4) partial per block.
__global__ void __launch_bounds__(TPB)
yolo_partial_kernel(const float* __restrict__ post,
                    const float* __restrict__ boxes,
                    float* __restrict__ partials, int n) {
    const int  tid    = blockIdx.x * TPB + threadIdx.x;
    const long stride = (long)gridDim.x * TPB;

    float accL0 = 0.0f, accL1 = 0.0f;   // loss partials (feed WMMA A slots)
    float accS0 = 0.0f, accS1 = 0.0f;   // post[:,4] sum partials

    for (long i = tid; i < n; i += 2 * stride) {
        {
            // (conf, cls) pair: byte offset (6i+4)*4 = 24i+16, 8B aligned
            float2 cs = *(const float2*)(post + i * 6 + 4);
            float4 bx = *(const float4*)(boxes + i * 4);     // 16B aligned
            float  m  = (cs.x >= CONF_THRES) ? 1.0f : 0.0f;
            accL0 += m * (cs.y + ((bx.x + bx.y) + (bx.z + bx.w)));
            accS0 += cs.x;
        }
        long j = i + stride;
        if (j < n) {
            float2 cs = *(const float2*)(post + j * 6 + 4);
            float4 bx = *(const float4*)(boxes + j * 4);
            float  m  = (cs.x >= CONF_THRES) ? 1.0f : 0.0f;
            accL1 += m * (cs.y + ((bx.x + bx.y) + (bx.z + bx.w)));
            accS1 += cs.x;
        }
    }

    // All lanes reconverged here: EXEC all-ones for WMMA.
    float waveL = wave_sum_f32_wmma(accL0, accL1);
    float waveS = wave_sum_f32_wmma(accS0, accS1);

    __shared__ float sL[NWAVES];
    __shared__ float sS[NWAVES];
    const int wave = threadIdx.x >> 5;
    const int lane = threadIdx.x & 31;
    if (lane == 0) { sL[wave] = waveL; sS[wave] = waveS; }
    __syncthreads();

    if (threadIdx.x == 0) {
        float tl = 0.0f, ts = 0.0f;
        #pragma unroll
        for (int w = 0; w < NWAVES; ++w) { tl += sL[w]; ts += sS[w]; }
        partials[blockIdx.x]        = tl;
        partials[NBLK + blockIdx.x] = ts;
    }
}

// Pass 2: deterministic reduction of the NBLK partials + fallback semantics.
__global__ void __launch_bounds__(TPB)
yolo_final_kernel(const float* __restrict__ partials, float* __restrict__ out) {
    const int tid = threadIdx.x;

    float aL0 = 0.0f, aL1 = 0.0f, aS0 = 0.0f, aS1 = 0.0f;
    // NBLK=1024, TPB=256: all iterations fully uniform (no divergence).
    #pragma unroll
    for (int i = tid; i < NBLK; i += 2 * TPB) {
        aL0 += partials[i];
        aS0 += partials[NBLK + i];
        int j = i + TPB;
        aL1 += partials[j];
        aS1 += partials[NBLK + j];
    }

    float waveL = wave_sum_f32_wmma(aL0, aL1);
    float waveS = wave_sum_f32_wmma(aS0, aS1);

    __shared__ float sL[NWAVES];
    __shared__ float sS[NWAVES];
    const int wave = threadIdx.x >> 5;
    const int lane = threadIdx.x & 31;
    if (lane == 0) { sL[wave] = waveL; sS[wave] = waveS; }
    __syncthreads();

    if (threadIdx.x == 0) {
        float loss = 0.0f, s4 = 0.0f;
        #pragma unroll
        for (int w = 0; w < NWAVES; ++w) { loss += sL[w]; s4 += sS[w]; }
        // reference fallback: loss == 0.0 -> post[:,4].sum() * 0.0
        out[0] = (loss == 0.0f) ? (s4 * 0.0f) : loss;
    }
}

extern "C" void kernel_launch(void* const* d_in, const int* in_sizes, int n_in,
                              void* d_out, int out_size, void* d_ws, size_t ws_size,
                              hipStream_t stream) {
    const float* post  = (const float*)d_in[0];   // [N,6] f32
    const float* boxes = (const float*)d_in[1];   // [N,4] f32
    const int n = in_sizes[0] / 6;

    float* partials = (float*)d_ws;               // 2*NBLK floats = 8 KB

    yolo_partial_kernel<<<NBLK, TPB, 0, stream>>>(post, boxes, partials, n);
    yolo_final_kernel<<<1, TPB, 0, stream>>>(partials, (float*)d_out);
}